// AdjMatrixGenerator_61211873903191
// MI455X (gfx1250) — compile-verified
//
#include <hip/hip_runtime.h>
#include <math.h>

typedef float v2f __attribute__((ext_vector_type(2)));
typedef float v8f __attribute__((ext_vector_type(8)));

#define BB 16
#define NN 256
#define FF 64
#define HH 64

// ---------------------------------------------------------------------------
// Stage 1: a[b,n,h] = sum_f X[b,n,f]*w1[h,f]       (+ b1[h] folded in)
//          c[b,n,h] = sum_f X[b,n,f]*w1[h,F+f]
// One wave (32 threads) per 16x16 output tile, K=64 via 16x V_WMMA_F32_16X16X4_F32.
// Column space p in [0,128): p<64 -> a-part (h=p), p>=64 -> c-part (h=p-64).
// ---------------------------------------------------------------------------
__global__ __launch_bounds__(32)
void gemm_stage1(const float* __restrict__ X,   // (B*N, F) = (4096, 64)
                 const float* __restrict__ W1,  // (H, 2F)  = (64, 128)
                 const float* __restrict__ B1,  // (H,)
                 float* __restrict__ a_out,     // (4096, 64), includes +b1
                 float* __restrict__ c_out)     // (4096, 64)
{
    const int lane  = threadIdx.x;    // 0..31, full wave, EXEC all-ones
    const int half  = lane >> 4;      // 0 or 1
    const int l     = lane & 15;
    const int tile_m = blockIdx.x;    // 0..255  (row tiles of 16)
    const int tile_p = blockIdx.y;    // 0..7    (column tiles of 16 over p=0..127)

    const int col_off = (tile_p >= 4) ? FF : 0;   // which half of w1
    const int h_base  = (tile_p & 3) * 16;        // h tile base

    // A fragment rows: M = l; VGPR0 holds K = 2*half, VGPR1 holds K = 2*half+1
    const float* xrow = X + (tile_m * 16 + l) * FF;
    // B fragment: B[k, n] = w1[h_base+n, col_off+k]; N = l striped across lanes
    const float* wrow = W1 + (h_base + l) * (2 * FF) + col_off;

    v8f acc = {};
    #pragma unroll
    for (int k = 0; k < FF; k += 4) {
        v2f a, b;
        a.x = xrow[k + 2 * half + 0];
        a.y = xrow[k + 2 * half + 1];
        b.x = wrow[k + 2 * half + 0];
        b.y = wrow[k + 2 * half + 1];
        // 8 args: (neg_a, A, neg_b, B, c_mod, C, reuse_a, reuse_b)
        acc = __builtin_amdgcn_wmma_f32_16x16x4_f32(
            false, a, false, b, (short)0, acc, false, false);
    }

    // D layout: vreg v -> M = v + 8*half, N = l  (h = h_base + l)
    const int h = h_base + l;
    const bool is_a = (tile_p < 4);
    const float bias = is_a ? B1[h] : 0.0f;
    #pragma unroll
    for (int v = 0; v < 8; ++v) {
        const int orow = tile_m * 16 + v + 8 * half;
        if (is_a) a_out[orow * HH + h] = acc[v] + bias;
        else      c_out[orow * HH + h] = acc[v];
    }
}

// ---------------------------------------------------------------------------
// Stage 2: adj[b,i,j] = sigmoid( sum_h relu(a'[b,i,h] + c[b,j,h]) * w2[h] + b2 )
//          diagonal forced to 1.
// 256-thread block computes a 32x32 (i,j) tile; each thread owns a 2x2
// micro-tile so every 4 LDS b128 reads feed 4 accumulators (12:1 VALU:DS,
// LDS no longer the bottleneck). Row stride 68 floats: float4-aligned,
// conflict-free banks for the strided tj reads.
// ---------------------------------------------------------------------------
__global__ __launch_bounds__(256)
void pairwise_stage2(const float* __restrict__ a_ws,  // (4096, 64), has +b1
                     const float* __restrict__ c_ws,  // (4096, 64)
                     const float* __restrict__ w2,    // (64,)
                     const float* __restrict__ b2,    // (1,)
                     float* __restrict__ out)         // (16, 256, 256)
{
    constexpr int LSTR = HH + 4;  // 68 floats
    __shared__ float sa[32 * LSTR];
    __shared__ float sc[32 * LSTR];
    __shared__ float sw[HH];

    const int tid = threadIdx.x;
    const int b   = blockIdx.z;
    const int i0  = blockIdx.y * 32;
    const int j0  = blockIdx.x * 32;

    // Stage 32x64 a-tile and 32x64 c-tile into LDS (float4 per load, coalesced)
    for (int idx = tid; idx < 32 * (HH / 4); idx += 256) {
        const int r = idx >> 4;          // row in tile (0..31)
        const int q = (idx & 15) * 4;    // h offset (0,4,...,60)
        *(float4*)(&sa[r * LSTR + q]) = *(const float4*)(&a_ws[((b * NN + i0 + r) * HH) + q]);
        *(float4*)(&sc[r * LSTR + q]) = *(const float4*)(&c_ws[((b * NN + j0 + r) * HH) + q]);
    }
    if (tid < HH) sw[tid] = w2[tid];
    __syncthreads();

    const int ti = tid >> 4;   // 0..15 -> rows ti, ti+16
    const int tj = tid & 15;   // 0..15 -> cols tj, tj+16
    const float* pa0 = &sa[ti * LSTR];
    const float* pa1 = &sa[(ti + 16) * LSTR];
    const float* pc0 = &sc[tj * LSTR];
    const float* pc1 = &sc[(tj + 16) * LSTR];

    float acc00 = 0.0f, acc01 = 0.0f, acc10 = 0.0f, acc11 = 0.0f;
    #pragma unroll
    for (int h = 0; h < HH; h += 4) {
        const float4 va0 = *(const float4*)(pa0 + h);
        const float4 va1 = *(const float4*)(pa1 + h);
        const float4 vc0 = *(const float4*)(pc0 + h);
        const float4 vc1 = *(const float4*)(pc1 + h);
        const float w0 = sw[h + 0], w1 = sw[h + 1], w2v = sw[h + 2], w3 = sw[h + 3];

        acc00 += fmaxf(va0.x + vc0.x, 0.0f) * w0;
        acc00 += fmaxf(va0.y + vc0.y, 0.0f) * w1;
        acc00 += fmaxf(va0.z + vc0.z, 0.0f) * w2v;
        acc00 += fmaxf(va0.w + vc0.w, 0.0f) * w3;

        acc01 += fmaxf(va0.x + vc1.x, 0.0f) * w0;
        acc01 += fmaxf(va0.y + vc1.y, 0.0f) * w1;
        acc01 += fmaxf(va0.z + vc1.z, 0.0f) * w2v;
        acc01 += fmaxf(va0.w + vc1.w, 0.0f) * w3;

        acc10 += fmaxf(va1.x + vc0.x, 0.0f) * w0;
        acc10 += fmaxf(va1.y + vc0.y, 0.0f) * w1;
        acc10 += fmaxf(va1.z + vc0.z, 0.0f) * w2v;
        acc10 += fmaxf(va1.w + vc0.w, 0.0f) * w3;

        acc11 += fmaxf(va1.x + vc1.x, 0.0f) * w0;
        acc11 += fmaxf(va1.y + vc1.y, 0.0f) * w1;
        acc11 += fmaxf(va1.z + vc1.z, 0.0f) * w2v;
        acc11 += fmaxf(va1.w + vc1.w, 0.0f) * w3;
    }

    const float bias2 = b2[0];
    const int i_a = i0 + ti, i_b = i0 + ti + 16;
    const int j_a = j0 + tj, j_b = j0 + tj + 16;

    const float s00 = 1.0f / (1.0f + __expf(-(acc00 + bias2)));
    const float s01 = 1.0f / (1.0f + __expf(-(acc01 + bias2)));
    const float s10 = 1.0f / (1.0f + __expf(-(acc10 + bias2)));
    const float s11 = 1.0f / (1.0f + __expf(-(acc11 + bias2)));

    float* outb = out + (size_t)b * NN * NN;
    outb[i_a * NN + j_a] = (i_a == j_a) ? 1.0f : s00;
    outb[i_a * NN + j_b] = (i_a == j_b) ? 1.0f : s01;
    outb[i_b * NN + j_a] = (i_b == j_a) ? 1.0f : s10;
    outb[i_b * NN + j_b] = (i_b == j_b) ? 1.0f : s11;
}

extern "C" void kernel_launch(void* const* d_in, const int* in_sizes, int n_in,
                              void* d_out, int out_size, void* d_ws, size_t ws_size,
                              hipStream_t stream) {
    const float* X   = (const float*)d_in[0];  // hidden_state (16,256,64)
    const float* W1  = (const float*)d_in[1];  // w1 (64,128)
    const float* B1v = (const float*)d_in[2];  // b1 (64,)
    const float* W2  = (const float*)d_in[3];  // w2 (1,64)
    const float* B2v = (const float*)d_in[4];  // b2 (1,)
    float* out = (float*)d_out;

    float* a_ws = (float*)d_ws;                 // 4096*64 floats = 1 MB
    float* c_ws = a_ws + (size_t)BB * NN * HH;  // next 1 MB

    dim3 g1(BB * NN / 16, 8);   // 256 row tiles x 8 column tiles (p in [0,128))
    gemm_stage1<<<g1, 32, 0, stream>>>(X, W1, B1v, a_ws, c_ws);

    dim3 g2(NN / 32, NN / 32, BB);
    pairwise_stage2<<<g2, 256, 0, stream>>>(a_ws, c_ws, W2, B2v, out);
}